// Quanvolution__gen364_65481071400243
// MI455X (gfx1250) — compile-verified
//
#include <hip/hip_runtime.h>

// ---------------------------------------------------------------------------
// Quanvolution fused kernel for gfx1250 (CDNA5, wave32).
//  Phase 0: stage zero-padded W [16x800] into LDS (coalesced b128)
//  Phase 1: per-patch 4-qubit statevector sim (pure VALU, in registers)
//  Phase 2: feats[16x800] @ W.T[800x16] via V_WMMA_F32_16X16X4_F32,
//           K padded to 800 so each of the 8 waves does exactly 25 K-steps
//           (uniform literal trip count -> fully unrolled, no EXEC masking)
//  Phase 3: reduce partials, bias + log_softmax
// ---------------------------------------------------------------------------

typedef float v2f __attribute__((ext_vector_type(2)));
typedef float v8f __attribute__((ext_vector_type(8)));

#define ROWS 16          // images per block (one WMMA M-tile)
#define KDIM 784         // 196 patches * 4 features
#define KPAD 800         // 800 = 8 waves * 25 steps * K4 (cols 784..799 zeroed)
#define LDS_STRIDE 804   // 804*4 = 3216 B: 16B-aligned rows, 36*m%64 distinct banks

// RY(theta) butterfly on qubit q of a 16-amplitude real state.
// Basis index = b0*8 + b1*4 + b2*2 + b3 (qubit w on axis w+1 of [N,2,2,2,2]).
__device__ __forceinline__ void apply_ry(float st[16], int q, float c, float s) {
  const int bit = 8 >> q;
#pragma unroll
  for (int i = 0; i < 16; ++i) {
    if (!(i & bit)) {
      const float a0 = st[i], a1 = st[i | bit];
      st[i]       = __builtin_fmaf(-s, a1, c * a0);
      st[i | bit] = __builtin_fmaf( s, a0, c * a1);
    }
  }
}

// CNOT = compile-time register permutation (zero VALU after renaming).
__device__ __forceinline__ void apply_cnot(float st[16], int cq, int tq) {
  const int cb = 8 >> cq, tb = 8 >> tq;
#pragma unroll
  for (int i = 0; i < 16; ++i) {
    if ((i & cb) && !(i & tb)) {
      const float t = st[i];
      st[i]      = st[i | tb];
      st[i | tb] = t;
    }
  }
}

__global__ __launch_bounds__(256)
void quanv_fused_kernel(const float* __restrict__ x,     // [B,1,28,28]
                        const float* __restrict__ wgt,   // [2,4]
                        const float* __restrict__ Wm,    // [10,784]
                        const float* __restrict__ bias,  // [10]
                        float* __restrict__ out,         // [B,10] log-softmax
                        int B) {
  __shared__ __align__(16) float sA[ROWS][LDS_STRIDE];   // feats tile (f32)
  __shared__ __align__(16) float sB[16][LDS_STRIDE];     // W zero-padded to 16x800
  __shared__ __align__(16) float sC[8][16][16];          // per-wave partial C

  const int tid  = threadIdx.x;
  const int row0 = blockIdx.x * ROWS;

  // ---------------- Phase 0: stage W into LDS (rows 10..15 + cols >=784 zero)
  if (tid < KPAD / 4) {                       // 200 threads, float4 granularity
#pragma unroll 1
    for (int n = 0; n < 16; ++n) {
      float4 v = make_float4(0.f, 0.f, 0.f, 0.f);
      if (n < 10 && tid < KDIM / 4) v = *(const float4*)(Wm + n * KDIM + 4 * tid);
      *(float4*)&sB[n][4 * tid] = v;
    }
  }
  // Zero the K-pad tail of the feats tile: 16 rows x 16 cols = 256 elems.
  sA[tid >> 4][KDIM + (tid & 15)] = 0.0f;

  // Variational-layer sin/cos: 8 uniform values, hoisted out of the patch loop.
  float wc[8], ws[8];
#pragma unroll
  for (int k = 0; k < 8; ++k) __sincosf(wgt[k] * 0.5f, &ws[k], &wc[k]);

  // ---------------- Phase 1: quantum sim -> feats in LDS ----------------
  for (int it = tid; it < ROWS * 196; it += 256) {
    const int r = it / 196;
    const int q = it - r * 196;       // patch id = i*14 + j
    const int i = q / 14;
    const int j = q - i * 14;
    if (row0 + r < B) {
      const float* xb = x + (size_t)(row0 + r) * 784;
      const v2f top = *(const v2f*)(xb + 56 * i + 2 * j);        // (i,j),(i,j+1)
      const v2f bot = *(const v2f*)(xb + 56 * i + 28 + 2 * j);   // (i+1,j),(i+1,j+1)

      float c0, s0, c1, s1, c2, s2, c3, s3;
      __sincosf(top.x * 0.5f, &s0, &c0);
      __sincosf(top.y * 0.5f, &s1, &c1);
      __sincosf(bot.x * 0.5f, &s2, &c2);
      __sincosf(bot.y * 0.5f, &s3, &c3);

      // Encoding on |0000> yields a product state: build it with 24 mults.
      float a01[4] = {c0 * c1, c0 * s1, s0 * c1, s0 * s1};
      float a23[4] = {c2 * c3, c2 * s3, s2 * c3, s2 * s3};
      float st[16];
#pragma unroll
      for (int u = 0; u < 4; ++u)
#pragma unroll
        for (int v = 0; v < 4; ++v) st[u * 4 + v] = a01[u] * a23[v];

      // Two variational layers.
#pragma unroll
      for (int d = 0; d < 2; ++d) {
#pragma unroll
        for (int qq = 0; qq < 4; ++qq)
          apply_ry(st, qq, wc[d * 4 + qq], ws[d * 4 + qq]);
        apply_cnot(st, 0, 1);
        apply_cnot(st, 1, 2);
        apply_cnot(st, 2, 3);
      }

      // <Z_q> = sum_{bit_q=0} p - sum_{bit_q=1} p
      float p[16];
#pragma unroll
      for (int u = 0; u < 16; ++u) p[u] = st[u] * st[u];
      float z[4];
#pragma unroll
      for (int qq = 0; qq < 4; ++qq) {
        const int bit = 8 >> qq;
        float acc = 0.0f;
#pragma unroll
        for (int u = 0; u < 16; ++u) acc += (u & bit) ? -p[u] : p[u];
        z[qq] = acc;
      }
      *(float4*)&sA[r][4 * q] = make_float4(z[0], z[1], z[2], z[3]);
    }
  }

  __syncthreads();

  // ---------------- Phase 2: WMMA GEMM, K split across all 8 waves ------
  // 200 K-steps of 4 over the padded K=800; each wave does exactly 25 steps.
  {
    const int wave   = tid >> 5;
    const int lane   = tid & 31;
    const int n      = lane & 15;   // N column; A row index m == lane&15 too
    const int hi     = lane >> 4;   // half-wave selects K pair {0,1} vs {2,3}
    const int kstart = wave * 100 + hi * 2;

    v8f c = {};
#pragma unroll
    for (int s = 0; s < 25; ++s) {
      const int k0 = kstart + 4 * s;
      // A 16x4 f32 frag: lane m holds feats[m][k0], feats[m][k0+1]
      const v2f a = *(const v2f*)&sA[n][k0];
      // B 4x16 f32 frag: lane n holds W[n][k0], W[n][k0+1] (pad rows/cols zero)
      const v2f b = *(const v2f*)&sB[n][k0];
      c = __builtin_amdgcn_wmma_f32_16x16x4_f32(
          /*neg_a=*/false, a, /*neg_b=*/false, b,
          /*c_mod=*/(short)0, c, /*reuse_a=*/false, /*reuse_b=*/false);
    }
    // C 16x16 f32: VGPR r -> M=r (lanes 0-15) / M=r+8 (lanes 16-31), N=lane&15
#pragma unroll
    for (int r = 0; r < 8; ++r) sC[wave][r + hi * 8][n] = c[r];
  }

  __syncthreads();

  // ---------------- Phase 3: reduce partials, bias, log_softmax ---------
  if (tid < ROWS && row0 + tid < B) {
    float l[10];
    float mx = -1e30f;
#pragma unroll
    for (int nn = 0; nn < 10; ++nn) {
      float v = bias[nn];
#pragma unroll
      for (int w = 0; w < 8; ++w) v += sC[w][tid][nn];
      l[nn] = v;
      mx = fmaxf(mx, v);
    }
    float sum = 0.0f;
#pragma unroll
    for (int nn = 0; nn < 10; ++nn) sum += __expf(l[nn] - mx);
    const float lse = __logf(sum);
    float* po = out + (size_t)(row0 + tid) * 10;
#pragma unroll
    for (int nn = 0; nn < 10; ++nn) po[nn] = l[nn] - mx - lse;
  }
}

extern "C" void kernel_launch(void* const* d_in, const int* in_sizes, int n_in,
                              void* d_out, int out_size, void* d_ws, size_t ws_size,
                              hipStream_t stream) {
  const float* x    = (const float*)d_in[0];  // [B,1,28,28]
  const float* wgt  = (const float*)d_in[1];  // [2,4]
  const float* Wm   = (const float*)d_in[2];  // [10,784]
  const float* bias = (const float*)d_in[3];  // [10]
  float* out = (float*)d_out;

  const int B = in_sizes[0] / 784;            // 8192
  const int grid = (B + ROWS - 1) / ROWS;     // 512 blocks of 256 threads
  quanv_fused_kernel<<<grid, 256, 0, stream>>>(x, wgt, Wm, bias, out, B);
}